// MoEGate_66967130079939
// MI455X (gfx1250) — compile-verified
//
#include <hip/hip_runtime.h>
#include <hip/hip_bf16.h>

typedef float v2f __attribute__((ext_vector_type(2)));
typedef float v8f __attribute__((ext_vector_type(8)));

#define S_SEQ 8192
#define BB 8
#define DDIM 1024
#define NE 31
#define KTOP 3
#define T_ROWS (S_SEQ * BB)          // 65536 rows
#define ROWS_PER_BLOCK 128
#define BK 64                        // k-chunk staged in LDS
#define NCHUNK (DDIM / BK)           // 16
#define XSTRIDE 68                   // 64 + 4 pad -> conflict-free strided A reads
#define LSTRIDE 33                   // logits row stride (32 + 1)
#define NBLOCKS (T_ROWS / ROWS_PER_BLOCK)   // 512
#define PART_STRIDE (2 * NE)         // per-block partials: [score-sum x31][count x31]

// Issue one 128x64 fp32 x-tile chunk as 8 async global->LDS B128 DMAs (per thread).
// ASYNCcnt-tracked; no VGPR round-trip, overlaps with WMMA on the other buffer.
__device__ __forceinline__ void issue_chunk_async(
    const float* __restrict__ gRowPtr,  // x + (rowBase + ldRow)*D + ldCol
    float* ldsBase,                     // &xbuf[buf][0]
    int ldRow, int ldCol, int kc)
{
    #pragma unroll
    for (int p = 0; p < 8; ++p) {
        unsigned lds = (unsigned)(uintptr_t)(ldsBase + (ldRow + p * 16) * XSTRIDE + ldCol);
        const float* ga = gRowPtr + (long)p * 16 * DDIM + kc * BK;
        asm volatile("global_load_async_to_lds_b128 %0, %1, off"
                     :: "v"(lds), "v"(ga) : "memory");
    }
}

__global__ __launch_bounds__(256) void moe_gate_kernel(
    const float* __restrict__ x, const float* __restrict__ w,
    float* __restrict__ out, float* __restrict__ parts)
{
    __shared__ __align__(16) float xbuf[2][ROWS_PER_BLOCK * XSTRIDE]; // 2 x 34.8 KB
    __shared__ __align__(16) float lg[ROWS_PER_BLOCK * LSTRIDE];      // 16.9 KB
    __shared__ float cnt[NE];

    const int tid   = threadIdx.x;
    const int wave  = tid >> 5;          // 0..7, each wave owns 16 rows
    const int lane  = tid & 31;
    const int lhalf = lane >> 4;         // 0/1
    const int llow  = lane & 15;
    const int blk   = blockIdx.x;
    const long rowBase = (long)blk * ROWS_PER_BLOCK;

    if (tid < NE) cnt[tid] = 0.0f;

    v8f acc0 = {};   // experts 0..15
    v8f acc1 = {};   // experts 16..30 (col 31 = harmless garbage, never read)

    // B-fragment base pointers (weight is (E, D) row-major; B[k][n] = w[n][k]).
    // Expert 31 is clamped to a valid address; its output column is ignored.
    const float* wp0 = w + llow * DDIM + 2 * lhalf;
    const int n1 = 16 + llow;
    const float* wp1 = w + (n1 < NE ? n1 : 0) * DDIM + 2 * lhalf;

    const int ldRow = tid >> 4;          // 0..15
    const int ldCol = (tid & 15) * 4;    // 0..60 (float4 column)
    const float* gRowPtr = x + (rowBase + ldRow) * DDIM + ldCol;

    // Prologue: DMA chunk 0, wait, barrier.
    issue_chunk_async(gRowPtr, &xbuf[0][0], ldRow, ldCol, 0);
    asm volatile("s_wait_asynccnt 0x0" ::: "memory");
    __syncthreads();

    for (int kc = 0; kc < NCHUNK; ++kc) {
        // Kick off next chunk's DMA into the other buffer (WAR-safe: the
        // barrier that ended iteration kc-1 retired all reads of that buffer).
        if (kc + 1 < NCHUNK)
            issue_chunk_async(gRowPtr, &xbuf[(kc + 1) & 1][0], ldRow, ldCol, kc + 1);

        const float* xb = &xbuf[kc & 1][(wave * 16 + llow) * XSTRIDE + 2 * lhalf];
        const int kbase = kc * BK;
        #pragma unroll
        for (int kk = 0; kk < BK; kk += 4) {
            // A (16x4 f32): lanes 0-15 -> K=kk,kk+1 ; lanes 16-31 -> K=kk+2,kk+3
            v2f a  = *(const v2f*)(xb + kk);
            v2f b0 = *(const v2f*)(wp0 + kbase + kk);   // cached weight reads
            v2f b1 = *(const v2f*)(wp1 + kbase + kk);
            acc0 = __builtin_amdgcn_wmma_f32_16x16x4_f32(false, a, false, b0,
                                                         (short)0, acc0, false, false);
            acc1 = __builtin_amdgcn_wmma_f32_16x16x4_f32(false, a, false, b1,
                                                         (short)0, acc1, false, false);
        }
        // Next chunk's DMA must have landed before anyone reads it.
        asm volatile("s_wait_asynccnt 0x0" ::: "memory");
        __syncthreads();
    }

    // Scatter logits to LDS: C/D layout -> VGPR r, lanes0-15: M=r, lanes16-31: M=r+8
    #pragma unroll
    for (int r = 0; r < 8; ++r) {
        const int m = wave * 16 + r + 8 * lhalf;
        lg[m * LSTRIDE + llow]      = acc0[r];
        lg[m * LSTRIDE + 16 + llow] = acc1[r];
    }
    __syncthreads();

    // Per-row softmax + top-3 (threads 0..127, one row each)
    if (tid < ROWS_PER_BLOCK) {
        float l[NE];
        float mx = -3.4e38f;
        #pragma unroll
        for (int e = 0; e < NE; ++e) { l[e] = lg[tid * LSTRIDE + e]; mx = fmaxf(mx, l[e]); }
        float sum = 0.0f;
        #pragma unroll
        for (int e = 0; e < NE; ++e) { l[e] = __expf(l[e] - mx); sum += l[e]; }
        const float inv = 1.0f / sum;
        #pragma unroll
        for (int e = 0; e < NE; ++e) l[e] *= inv;   // softmax scores

        // top-3, first-occurrence-wins on ties (matches jax.lax.top_k)
        int i0 = 0; float s0 = l[0];
        #pragma unroll
        for (int e = 1; e < NE; ++e) if (l[e] > s0) { s0 = l[e]; i0 = e; }
        int i1 = -1; float s1 = -1.0f;
        #pragma unroll
        for (int e = 0; e < NE; ++e) if (e != i0 && l[e] > s1) { s1 = l[e]; i1 = e; }
        int i2 = -1; float s2 = -1.0f;
        #pragma unroll
        for (int e = 0; e < NE; ++e) if (e != i0 && e != i1 && l[e] > s2) { s2 = l[e]; i2 = e; }

        const float invd = 1.0f / (s0 + s1 + s2 + 1e-20f);
        const long t = rowBase + tid;
        float* oi = out;                       // indices as float values
        float* ow = out + (long)3 * T_ROWS;    // normalized weights
        oi[t * 3 + 0] = (float)i0;
        oi[t * 3 + 1] = (float)i1;
        oi[t * 3 + 2] = (float)i2;
        ow[t * 3 + 0] = s0 * invd;
        ow[t * 3 + 1] = s1 * invd;
        ow[t * 3 + 2] = s2 * invd;

        // aux-loss stats: counts (exact integer fp32 adds -> deterministic)
        atomicAdd(&cnt[i0], 1.0f);
        atomicAdd(&cnt[i1], 1.0f);
        atomicAdd(&cnt[i2], 1.0f);
        // write scores back for deterministic column reduction
        #pragma unroll
        for (int e = 0; e < NE; ++e) lg[tid * LSTRIDE + e] = l[e];
    }
    __syncthreads();

    // Per-block partials (no global float atomics -> bitwise deterministic)
    if (tid < NE) {
        float s = 0.0f;
        for (int r = 0; r < ROWS_PER_BLOCK; ++r) s += lg[r * LSTRIDE + tid];
        parts[blk * PART_STRIDE + tid]      = s;         // score sum
        parts[blk * PART_STRIDE + NE + tid] = cnt[tid];  // top-k count
    }
}

__global__ __launch_bounds__(256) void moe_aux_kernel(
    const float* __restrict__ parts, float* __restrict__ out)
{
    __shared__ float terms[BB * NE];
    const int j = threadIdx.x;
    if (j < BB * NE) {
        const int b = j / NE, e = j % NE;
        float s = 0.0f, c = 0.0f;
        const int bpg = NBLOCKS / BB;   // 64 blocks per sequence-group
        for (int i = 0; i < bpg; ++i) {
            const int blkI = b * bpg + i;
            s += parts[blkI * PART_STRIDE + e];
            c += parts[blkI * PART_STRIDE + NE + e];
        }
        const float ce = c * ((float)NE / (float)(S_SEQ * KTOP));
        terms[j] = ce * (s / (float)S_SEQ);
    }
    __syncthreads();
    if (j == 0) {
        float loss = 0.0f;
        for (int i = 0; i < BB * NE; ++i) loss += terms[i];  // fixed order
        out[(long)6 * T_ROWS] = loss * (0.001f / (float)BB);
    }
}

extern "C" void kernel_launch(void* const* d_in, const int* in_sizes, int n_in,
                              void* d_out, int out_size, void* d_ws, size_t ws_size,
                              hipStream_t stream) {
    const float* x = (const float*)d_in[0];   // (8192, 8, 1024) fp32
    const float* w = (const float*)d_in[1];   // (31, 1024) fp32
    float* out   = (float*)d_out;             // [3T idx | 3T weights | 1 aux]
    float* parts = (float*)d_ws;              // 512 * 62 floats = 124 KB scratch

    moe_gate_kernel<<<NBLOCKS, 256, 0, stream>>>(x, w, out, parts);
    moe_aux_kernel<<<1, 256, 0, stream>>>(parts, out);
}